// vanillaGCN_60730837565902
// MI455X (gfx1250) — compile-verified
//
#include <hip/hip_runtime.h>

typedef __attribute__((ext_vector_type(2))) float v2f;
typedef __attribute__((ext_vector_type(8))) float v8f;

#define SA_STRIDE 66  // padded row stride (floats): lane-strided reads hit distinct banks, 8B aligned for even k

// ---------------- degree / normalization ----------------
__global__ void init_deg_kernel(float* deg, int n) {
    int t = blockIdx.x * blockDim.x + threadIdx.x;
    if (t < n) deg[t] = 1.0f;  // self-loop
}

__global__ void deg_accum_kernel(float* deg, const int* __restrict__ eidx, int E) {
    int e = blockIdx.x * blockDim.x + threadIdx.x;
    if (e < E) atomicAdd(&deg[eidx[E + e]], 1.0f);  // dst row of edge_index
}

__global__ void deg_to_dis_kernel(float* deg, int n) {
    int t = blockIdx.x * blockDim.x + threadIdx.x;
    if (t < n) deg[t] = rsqrtf(deg[t]);  // in place: deg -> deg^-1/2
}

// ---------------- zero fills ----------------
__global__ void zero_f4_kernel(float4* p, int n4) {
    int t = blockIdx.x * blockDim.x + threadIdx.x;
    if (t < n4) { float4 z = {0.f, 0.f, 0.f, 0.f}; p[t] = z; }
}

__global__ void zero_f_kernel(float* p, int n) {
    int t = blockIdx.x * blockDim.x + threadIdx.x;
    if (t < n) p[t] = 0.f;
}

// ---------------- GEMM: Out[nrows,64] = H[nrows,64] @ W[64,64] via V_WMMA_F32_16X16X4_F32 ----
// block = 256 threads = 8 waves; each wave owns one 16x16 tile; block covers 32 rows x 64 cols.
__global__ void __launch_bounds__(256)
gemm64_wmma_kernel(const float* __restrict__ H, const float* __restrict__ W,
                   float* __restrict__ Out, int nrows) {
    __shared__ float sW[64 * 64];
    __shared__ float sA[32 * SA_STRIDE];
    const int tid = threadIdx.x;

    for (int i = tid; i < 64 * 64; i += 256) sW[i] = W[i];

    const int rowbase = blockIdx.x * 32;
    for (int i = tid; i < 32 * 64; i += 256) {
        int r = i >> 6, cch = i & 63;
        int gr = rowbase + r;
        sA[r * SA_STRIDE + cch] = (gr < nrows) ? H[(size_t)gr * 64 + cch] : 0.f;
    }
    __syncthreads();

    const int wave = tid >> 5;
    const int lane = tid & 31;
    const int mt   = wave >> 2;            // row tile 0..1
    const int nt   = wave & 3;             // col tile 0..3
    const int lrow = lane & 15;            // M (A) / N (B,C,D) index
    const int khalf = (lane >> 4) << 1;    // lanes 0-15 -> K+{0,1}; lanes 16-31 -> K+{2,3}

    const float* aRow = &sA[(mt * 16 + lrow) * SA_STRIDE];
    const float* bCol = &sW[nt * 16 + lrow];

    v8f c = {};
#if __has_builtin(__builtin_amdgcn_wmma_f32_16x16x4_f32)
#pragma unroll
    for (int t = 0; t < 16; ++t) {
        int k = t * 4 + khalf;
        v2f a; a.x = aRow[k];        a.y = aRow[k + 1];
        v2f b; b.x = bCol[k * 64];   b.y = bCol[(k + 1) * 64];
        c = __builtin_amdgcn_wmma_f32_16x16x4_f32(false, a, false, b,
                                                  (short)0, c, false, false);
    }
#else
    // scalar fallback (compile robustness only): each lane computes its 8 C/D elements directly
    {
        int col = nt * 16 + lrow;
        int rofs = (lane >> 4) << 3;
#pragma unroll
        for (int r = 0; r < 8; ++r) {
            float acc = 0.f;
            const float* ar = &sA[(mt * 16 + rofs + r) * SA_STRIDE];
            for (int k = 0; k < 64; ++k) acc += ar[k] * sW[k * 64 + col];
            c[r] = acc;
        }
    }
#endif

    const int col  = nt * 16 + lrow;
    const int rofs = (lane >> 4) << 3;     // C/D: VGPR r holds M=r (lanes 0-15), M=r+8 (lanes 16-31)
#pragma unroll
    for (int r = 0; r < 8; ++r) {
        int row = rowbase + mt * 16 + rofs + r;
        if (row < nrows) Out[(size_t)row * 64 + col] = c[r];
    }
}

// ---------------- edge scatter: agg[dst] += hW[src] * dis[src]*dis[dst] ----------------
// 16 lanes per edge; each lane: one b128 load + 4 f32 atomics (L2-resident working set).
__global__ void edge_scatter_kernel(const float4* __restrict__ hw4,
                                    float* __restrict__ agg,
                                    const int* __restrict__ eidx,
                                    const float* __restrict__ dis, int E) {
    int t = blockIdx.x * blockDim.x + threadIdx.x;
    int e = t >> 4;
    int cch = t & 15;
    if (e >= E) return;
    int s = eidx[e];
    int d = eidx[E + e];
    float w = dis[s] * dis[d];
    float4 v = hw4[(size_t)s * 16 + cch];
    float* dp = agg + (size_t)d * 64 + cch * 4;
    atomicAdd(dp + 0, v.x * w);
    atomicAdd(dp + 1, v.y * w);
    atomicAdd(dp + 2, v.z * w);
    atomicAdd(dp + 3, v.w * w);
}

// ---------------- finalize: h = [relu](agg + self-loop + bias) ----------------
__global__ void finalize_kernel(const float* __restrict__ agg, const float* __restrict__ hw,
                                const float* __restrict__ dis, const float* __restrict__ bias,
                                float* __restrict__ hout, int relu, int n) {
    int t = blockIdx.x * blockDim.x + threadIdx.x;
    if (t >= n * 64) return;
    int i = t >> 6, f = t & 63;
    float di = dis[i];
    float v = agg[t] + hw[t] * di * di + bias[f];
    if (relu) v = fmaxf(v, 0.f);
    hout[t] = v;
}

// ---------------- pooling ----------------
__global__ void pool_accum_kernel(const float* __restrict__ h, const int* __restrict__ batch,
                                  float* __restrict__ pool, float* __restrict__ cnt, int n) {
    int t = blockIdx.x * blockDim.x + threadIdx.x;
    if (t >= n * 64) return;
    int i = t >> 6, f = t & 63;
    int g = batch[i];
    atomicAdd(&pool[g * 64 + f], h[t]);
    if (f == 0) atomicAdd(&cnt[g], 1.0f);
}

// one wave per graph: dot(pooled/cnt, lin_w) + lin_b
__global__ void readout_kernel(const float* __restrict__ pool, const float* __restrict__ cnt,
                               const float* __restrict__ lw, const float* __restrict__ lb,
                               float* __restrict__ out) {
    int g = blockIdx.x;
    int l = threadIdx.x;  // 0..31
    float s = pool[g * 64 + l] * lw[l] + pool[g * 64 + l + 32] * lw[l + 32];
    for (int o = 16; o > 0; o >>= 1) s += __shfl_xor(s, o, 32);
    if (l == 0) out[g] = s / fmaxf(cnt[g], 1.0f) + lb[0];
}

extern "C" void kernel_launch(void* const* d_in, const int* in_sizes, int n_in,
                              void* d_out, int out_size, void* d_ws, size_t ws_size,
                              hipStream_t stream) {
    const float* x     = (const float*)d_in[0];
    const int*   eidx  = (const int*)d_in[1];
    const int*   batch = (const int*)d_in[2];
    const float* Ws[4] = {(const float*)d_in[3], (const float*)d_in[5],
                          (const float*)d_in[7], (const float*)d_in[9]};
    const float* bs[4] = {(const float*)d_in[4], (const float*)d_in[6],
                          (const float*)d_in[8], (const float*)d_in[10]};
    const float* lw = (const float*)d_in[11];
    const float* lb = (const float*)d_in[12];
    float* out = (float*)d_out;

    const int N = in_sizes[0] / 64;
    const int E = in_sizes[1] / 2;
    const int G = out_size;  // N_PRED == 1

    // workspace layout (256B-aligned slices)
    char* ws = (char*)d_ws;
    size_t off = 0;
    auto carve = [&](size_t bytes) -> char* {
        char* p = ws + off;
        off = (off + bytes + 255) & ~(size_t)255;
        return p;
    };
    float* dis  = (float*)carve((size_t)N * 4);
    float* hW   = (float*)carve((size_t)N * 64 * 4);
    float* agg  = (float*)carve((size_t)N * 64 * 4);
    float* h    = (float*)carve((size_t)N * 64 * 4);
    float* pool = (float*)carve((size_t)G * 64 * 4);
    float* cnt  = (float*)carve((size_t)G * 4);
    (void)ws_size; (void)n_in;

    const int B = 256;
    // degrees -> dis = deg^-1/2 (self-loops included via init to 1)
    init_deg_kernel<<<(N + B - 1) / B, B, 0, stream>>>(dis, N);
    deg_accum_kernel<<<(E + B - 1) / B, B, 0, stream>>>(dis, eidx, E);
    deg_to_dis_kernel<<<(N + B - 1) / B, B, 0, stream>>>(dis, N);

    const float* hin = x;
    for (int l = 0; l < 4; ++l) {
        gemm64_wmma_kernel<<<(N + 31) / 32, 256, 0, stream>>>(hin, Ws[l], hW, N);
        zero_f4_kernel<<<((size_t)N * 16 + B - 1) / B, B, 0, stream>>>((float4*)agg, N * 16);
        edge_scatter_kernel<<<((size_t)E * 16 + B - 1) / B, B, 0, stream>>>(
            (const float4*)hW, agg, eidx, dis, E);
        finalize_kernel<<<((size_t)N * 64 + B - 1) / B, B, 0, stream>>>(
            agg, hW, dis, bs[l], h, (l < 3) ? 1 : 0, N);
        hin = h;
    }

    zero_f_kernel<<<(G * 65 + B - 1) / B, B, 0, stream>>>(pool, G * 65);  // pool + cnt contiguous
    pool_accum_kernel<<<((size_t)N * 64 + B - 1) / B, B, 0, stream>>>(h, batch, pool, cnt, N);
    readout_kernel<<<G, 32, 0, stream>>>(pool, cnt, lw, lb, out);
}